// PDFSampler_40415642255465
// MI455X (gfx1250) — compile-verified
//
#include <hip/hip_runtime.h>

#define NUM_RAYS 524288
#define ND 64
#define NS 32
#define PDF_EPS 1e-5f

#define WAVES_PER_BLOCK 4
#define RAYS_PER_WAVE 16
#define RAYS_PER_BLOCK (WAVES_PER_BLOCK * RAYS_PER_WAVE)   // 64
#define ROW_STRIDE 68   // floats; odd*4 -> conflict-free column gathers, 272B rows (16B aligned)
// per-wave LDS slice (floats):
//   [0,1088)      weights tile 16x68 (later overwritten in-place by raw cdf, 65 cols used)
//   [1088,2176)   s_offsets tile 16x68
//   [2176,2688)   s_new 16x32
//   [2688,2784)   96-float merge/output staging buffer
#define WAVE_SLICE 2784

typedef __attribute__((ext_vector_type(2))) float v2f;
typedef __attribute__((ext_vector_type(8))) float v8f;

__global__ __launch_bounds__(WAVES_PER_BLOCK * 32) void pdf_sample_kernel(
    const float* __restrict__ weights,
    const float* __restrict__ s_offsets,
    float* __restrict__ out)
{
    __shared__ __align__(16) float lds[WAVES_PER_BLOCK * WAVE_SLICE];

    const int lane = threadIdx.x & 31;
    const int wave = threadIdx.x >> 5;
    float* wlds = &lds[wave * WAVE_SLICE];          // weights -> raw cdf
    float* olds = wlds + 16 * ROW_STRIDE;           // offsets
    float* snew = olds + 16 * ROW_STRIDE;           // new samples
    float* obuf = snew + 16 * NS;                   // merge staging

    const int h  = lane >> 4;    // lane half
    const int ln = lane & 15;

    // ================= Phase 0: async global -> LDS staging (ASYNCcnt path) ==========
    {
        // block-uniform 64-bit bases for SADDR; wave/ray selection folded into VADDR
        const float* wbase = weights  + (size_t)blockIdx.x * RAYS_PER_BLOCK * ND;
        const float* obase = s_offsets + (size_t)blockIdx.x * RAYS_PER_BLOCK * ND;
        const int row = lane >> 4;    // 0..1
        const int seg = lane & 15;    // 16B segment within a 256B ray row
#pragma unroll
        for (int i = 0; i < 8; ++i) {
            const int r = 2 * i + row;                                  // ray-in-tile 0..15
            const int goff = ((wave * RAYS_PER_WAVE + r) * ND + seg * 4) * 4; // bytes
            const unsigned lw = (unsigned)(size_t)(wlds + r * ROW_STRIDE + seg * 4);
            const unsigned lo = (unsigned)(size_t)(olds + r * ROW_STRIDE + seg * 4);
            asm volatile("global_load_async_to_lds_b128 %0, %1, %2"
                         :: "v"(lw), "v"(goff), "s"(wbase) : "memory");
            asm volatile("global_load_async_to_lds_b128 %0, %1, %2"
                         :: "v"(lo), "v"(goff), "s"(obase) : "memory");
        }
        asm volatile("s_wait_asynccnt 0" ::: "memory");
    }

    // ================= Phase 1: 64-wide cumsum for 16 rays via V_WMMA_F32_16X16X4_F32 =
    // cdf_raw(16x64) = W(16x64) x LowerTriOnes(64x64), tiled as 4 16x16 output tiles,
    // K accumulated in chunks of 4.  A layout: lane ln holds M=ln, v0=K(2h), v1=K(2h+1).
    v2f a[16];
#pragma unroll
    for (int j = 0; j < 16; ++j)
        a[j] = *(const v2f*)(wlds + ln * ROW_STRIDE + 4 * j + 2 * h);

    const v2f bones = {1.0f, 1.0f};
    v2f btri[4];   // diagonal K-chunk jr of a tile: B[k,p] = (4*jr + k <= p)
#pragma unroll
    for (int jr = 0; jr < 4; ++jr) {
        const int g0 = 4 * jr + 2 * h;
        btri[jr].x = (g0     <= ln) ? 1.0f : 0.0f;
        btri[jr].y = (g0 + 1 <= ln) ? 1.0f : 0.0f;
    }

#pragma unroll
    for (int c = 0; c < 4; ++c) {                    // output position tile: cols 16c..16c+15
        v8f acc = {0.f, 0.f, 0.f, 0.f, 0.f, 0.f, 0.f, 0.f};
#pragma unroll
        for (int j = 0; j < 4 * c; ++j)              // fully-covered K chunks: all-ones B
            acc = __builtin_amdgcn_wmma_f32_16x16x4_f32(
                false, a[j], false, bones, (short)0, acc, false, false);
#pragma unroll
        for (int jr = 0; jr < 4; ++jr)               // diagonal chunks: triangular B
            acc = __builtin_amdgcn_wmma_f32_16x16x4_f32(
                false, a[4 * c + jr], false, btri[jr], (short)0, acc, false, false);
        // D tile -> LDS raw cdf at column pos+1 (ray = v + 8h, pos = 16c + ln)
#pragma unroll
        for (int v = 0; v < 8; ++v)
            wlds[(v + 8 * h) * ROW_STRIDE + 16 * c + ln + 1] = acc[v];
    }
    if (h == 0) wlds[ln * ROW_STRIDE] = 0.0f;        // cdf[0] = 0 for all 16 rays

    // ================= Phase 2: inverse-CDF sampling (32 u's per ray, one ray/iter) ====
    const float u = ((float)lane + 0.5f) * (1.0f / 32.0f);
#pragma unroll 1
    for (int i = 0; i < RAYS_PER_WAVE; ++i) {
        const float* cdf = wlds + i * ROW_STRIDE;    // 65 raw entries, cdf[0]=0
        const float* off = olds + i * ROW_STRIDE;
        const float total = cdf[64];
        const float pad   = fmaxf(PDF_EPS - total, 0.0f);
        const float inv   = 1.0f / (total + pad);
        const float padS  = pad * (1.0f / 64.0f);
        // cnorm(k) = (cdf[k] + padS*k) * inv ; searchsorted side='right' over 65 entries
        int lo = 0, hi = 64;                         // invariant cnorm(lo) <= u < cnorm(hi)
#pragma unroll
        for (int s = 0; s < 6; ++s) {
            const int mid = (lo + hi) >> 1;
            const float cm = (cdf[mid] + padS * (float)mid) * inv;
            const bool le = (cm <= u);
            lo = le ? mid : lo;
            hi = le ? hi  : mid;
        }
        const int il = hi - 1;                       // hi in [1,64]
        const int ir = (hi < 64) ? hi : 63;
        const float cl = (cdf[il] + padS * (float)il) * inv;
        const float cr = (cdf[ir] + padS * (float)ir) * inv;
        const float ol = off[il];
        const float orr = off[ir];
        const float den = cr - cl;
        const float num = u - cl;
        // nan_to_num((u-cl)/(cr-cl)) then clip[0,1]: 0/0->0, x/0->+inf->1
        const float t = (den > 0.0f) ? fminf(fmaxf(num / den, 0.0f), 1.0f)
                                     : (num > 0.0f ? 1.0f : 0.0f);
        snew[i * NS + lane] = ol + t * (orr - ol);
    }

    // ================= Phase 3: merge two sorted lists (32 + 64 -> 96) by rank =========
    const size_t outBase = (size_t)(blockIdx.x * RAYS_PER_BLOCK + wave * RAYS_PER_WAVE) * 96;
#pragma unroll 1
    for (int i = 0; i < RAYS_PER_WAVE; ++i) {
        const float* off = olds + i * ROW_STRIDE;
        const float* sn  = snew + i * NS;
        // rank of s_new[lane] = lane + #{off < v}  (branchless bisection over 64)
        {
            const float v = sn[lane];
            int base = 0;
#pragma unroll
            for (int s = 32; s >= 1; s >>= 1)
                if (off[base + s - 1] < v) base += s;
            obuf[lane + base] = v;
        }
        // rank of off[j] = j + #{snew <= v}  (branchless bisection over 32), 2 per lane
#pragma unroll
        for (int q = 0; q < 2; ++q) {
            const int j = lane + 32 * q;
            const float v = off[j];
            int base = 0;
#pragma unroll
            for (int s = 16; s >= 1; s >>= 1)
                if (sn[base + s - 1] <= v) base += s;
            obuf[j + base] = v;
        }
        // coalesced write-out: 3 stride-32 stores per lane
        float* orow = out + outBase + (size_t)i * 96;
        orow[lane]      = obuf[lane];
        orow[lane + 32] = obuf[lane + 32];
        orow[lane + 64] = obuf[lane + 64];
    }
}

extern "C" void kernel_launch(void* const* d_in, const int* in_sizes, int n_in,
                              void* d_out, int out_size, void* d_ws, size_t ws_size,
                              hipStream_t stream) {
    (void)n_in; (void)out_size; (void)d_ws; (void)ws_size;
    const float* weights   = (const float*)d_in[0];
    const float* s_offsets = (const float*)d_in[1];
    float* out = (float*)d_out;
    const int rays = in_sizes[0] / ND;                 // 524288
    const int grid = rays / RAYS_PER_BLOCK;            // 8192 blocks of 128 threads
    pdf_sample_kernel<<<grid, WAVES_PER_BLOCK * 32, 0, stream>>>(weights, s_offsets, out);
}